// Attention_32650341384897
// MI455X (gfx1250) — compile-verified
//
#include <hip/hip_runtime.h>
#include <math.h>

typedef float v2f __attribute__((ext_vector_type(2)));
typedef float v4f __attribute__((ext_vector_type(4)));
typedef float v8f __attribute__((ext_vector_type(8)));

#define B_    16
#define DEC_  2048
#define ENC_  2048
#define H_    1024
#define WAVES 8
#define ENC_PER_WAVE (ENC_ / WAVES)   // 256
#define NT1   (ENC_PER_WAVE / 16)     // 16 score tiles per wave
#define H_PER_WAVE (H_ / WAVES)       // 128
#define NT2   (H_PER_WAVE / 16)       // 8 context tiles per wave
#define PSTRIDE 2052                  // row stride: 16B-aligned, bank stride 4 -> conflict-free b64/b128
#define LDS_FLOATS (16 * PSTRIDE + 2 * WAVES * 16)

__launch_bounds__(256, 1)
__global__ void attn_fused_kernel(const float* __restrict__ dec,
                                  const float* __restrict__ enc,
                                  float* __restrict__ ctx_out,
                                  float* __restrict__ attn_out) {
    extern __shared__ float smem[];
    float* ldsP = smem;                       // 16 x PSTRIDE normalized probs
    float* redM = smem + 16 * PSTRIDE;        // 8 waves x 16 rowmax partials
    float* redS = redM + WAVES * 16;          // 8 waves x 16 rowsum partials

    const int tid  = threadIdx.x;
    const int wave = tid >> 5;
    const int lane = tid & 31;
    const int half = lane >> 4;   // 0: rows 0-7 / K 0-1 ; 1: rows 8-15 / K 2-3
    const int l15  = lane & 15;

    const int b     = blockIdx.x >> 7;     // / (DEC_/16)
    const int dtile = blockIdx.x & 127;
    const int d0    = dtile * 16;

    const float* decB = dec + ((size_t)b * DEC_ + d0) * H_;
    const float* encB = enc + (size_t)b * ENC_ * H_;

    const int colbase = wave * ENC_PER_WAVE;

    // ---------------- GEMM1: S = D * E^T  (16 x 256 per wave) ----------------
    v8f acc[NT1];
    #pragma unroll
    for (int t = 0; t < NT1; ++t) acc[t] = v8f{};

    for (int k = 0; k < H_; k += 4) {
        // A fragment: row = l15, K = k + 2*half + {0,1}
        v2f a = *(const v2f*)(decB + (size_t)l15 * H_ + k + 2 * half);
        #pragma unroll
        for (int t = 0; t < NT1; ++t) {
            const int col = colbase + t * 16 + l15;
            // B fragment: N = l15 (enc col), K = k + 2*half + {0,1}
            v2f bf = *(const v2f*)(encB + (size_t)col * H_ + k + 2 * half);
            acc[t] = __builtin_amdgcn_wmma_f32_16x16x4_f32(
                false, a, false, bf, (short)0, acc[t], false, false);
        }
    }

    // ---------------- mask: enc[:,0] == 0 -> -inf ----------------
    #pragma unroll
    for (int t = 0; t < NT1; ++t) {
        const int col = colbase + t * 16 + l15;
        const bool m = (encB[(size_t)col * H_] == 0.0f);
        if (m) {
            #pragma unroll
            for (int j = 0; j < 8; ++j) acc[t][j] = -INFINITY;
        }
    }

    // ---------------- row max (softmax) ----------------
    float mx[8];
    #pragma unroll
    for (int j = 0; j < 8; ++j) {
        float v = acc[0][j];
        #pragma unroll
        for (int t = 1; t < NT1; ++t) v = fmaxf(v, acc[t][j]);
        #pragma unroll
        for (int s = 1; s <= 8; s <<= 1) v = fmaxf(v, __shfl_xor(v, s, 32));
        mx[j] = v;
    }
    if (l15 == 0) {
        #pragma unroll
        for (int j = 0; j < 8; ++j) redM[wave * 16 + half * 8 + j] = mx[j];
    }
    __syncthreads();
    float rmax[8];
    #pragma unroll
    for (int j = 0; j < 8; ++j) {
        float v = redM[half * 8 + j];
        #pragma unroll
        for (int w = 1; w < WAVES; ++w) v = fmaxf(v, redM[w * 16 + half * 8 + j]);
        rmax[j] = v;
    }

    // ---------------- exp + row sum ----------------
    float sm[8];
    #pragma unroll
    for (int j = 0; j < 8; ++j) sm[j] = 0.0f;
    #pragma unroll
    for (int t = 0; t < NT1; ++t) {
        #pragma unroll
        for (int j = 0; j < 8; ++j) {
            float e = __expf(acc[t][j] - rmax[j]);  // masked: exp(-inf) = 0
            acc[t][j] = e;
            sm[j] += e;
        }
    }
    #pragma unroll
    for (int j = 0; j < 8; ++j) {
        float v = sm[j];
        #pragma unroll
        for (int s = 1; s <= 8; s <<= 1) v += __shfl_xor(v, s, 32);
        sm[j] = v;
    }
    if (l15 == 0) {
        #pragma unroll
        for (int j = 0; j < 8; ++j) redS[wave * 16 + half * 8 + j] = sm[j];
    }
    __syncthreads();
    float rinv[8];
    #pragma unroll
    for (int j = 0; j < 8; ++j) {
        float v = redS[half * 8 + j];
        #pragma unroll
        for (int w = 1; w < WAVES; ++w) v += redS[w * 16 + half * 8 + j];
        rinv[j] = 1.0f / v;
    }

    // ---------------- normalize, stage P in LDS ----------------
    #pragma unroll
    for (int t = 0; t < NT1; ++t) {
        const int col = colbase + t * 16 + l15;
        #pragma unroll
        for (int j = 0; j < 8; ++j) {
            const int row = j + 8 * half;
            ldsP[row * PSTRIDE + col] = acc[t][j] * rinv[j];
        }
    }
    __syncthreads();

    // ---------------- coalesced attn output (b128 from LDS) ----------------
    // Each thread owns 128 consecutive-ish columns of one row; stores overlap GEMM2.
    {
        float* attnB = attn_out + ((size_t)b * DEC_ + d0) * ENC_;
        const int srow  = tid >> 4;          // 0..15
        const int scol0 = (tid & 15) * 4;    // lanes adjacent -> contiguous 256B per 16 lanes
        #pragma unroll 4
        for (int i = 0; i < ENC_ / 64; ++i) {  // 32 iterations
            const int c = scol0 + i * 64;
            v4f p4 = *(const v4f*)&ldsP[srow * PSTRIDE + c];
            *(v4f*)&attnB[(size_t)srow * ENC_ + c] = p4;
        }
    }

    // ---------------- GEMM2: C = P * E  (16 x 128 per wave) ----------------
    const int hbase = wave * H_PER_WAVE;
    v8f cacc[NT2];
    #pragma unroll
    for (int t = 0; t < NT2; ++t) cacc[t] = v8f{};

    for (int ke = 0; ke < ENC_; ke += 4) {
        // A fragment from LDS P: row = l15, K(enc) = ke + 2*half + {0,1}
        v2f a = *(const v2f*)&ldsP[l15 * PSTRIDE + ke + 2 * half];
        #pragma unroll
        for (int t = 0; t < NT2; ++t) {
            const int hcol = hbase + t * 16 + l15;
            v2f bf;
            bf.x = encB[(size_t)(ke + 2 * half)     * H_ + hcol];
            bf.y = encB[(size_t)(ke + 2 * half + 1) * H_ + hcol];
            cacc[t] = __builtin_amdgcn_wmma_f32_16x16x4_f32(
                false, a, false, bf, (short)0, cacc[t], false, false);
        }
    }

    float* ctxB = ctx_out + ((size_t)b * DEC_ + d0) * H_;
    #pragma unroll
    for (int t = 0; t < NT2; ++t) {
        const int hcol = hbase + t * 16 + l15;
        #pragma unroll
        for (int j = 0; j < 8; ++j) {
            const int row = j + 8 * half;
            ctxB[(size_t)row * H_ + hcol] = cacc[t][j];
        }
    }
}

extern "C" void kernel_launch(void* const* d_in, const int* in_sizes, int n_in,
                              void* d_out, int out_size, void* d_ws, size_t ws_size,
                              hipStream_t stream) {
    const float* dec = (const float*)d_in[0];
    const float* enc = (const float*)d_in[1];
    float* ctx  = (float*)d_out;
    float* attn = (float*)d_out + (size_t)B_ * DEC_ * H_;

    const dim3 grid(B_ * (DEC_ / 16));   // 2048 workgroups
    const dim3 block(256);               // 8 waves (wave32)
    const size_t shmem = LDS_FLOATS * sizeof(float);  // ~132 KB of the 320 KB WGP LDS

    attn_fused_kernel<<<grid, block, shmem, stream>>>(dec, enc, ctx, attn);
}